// FeatureExtractor_79714593014084
// MI455X (gfx1250) — compile-verified
//
#include <hip/hip_runtime.h>

typedef __attribute__((ext_vector_type(2))) float v2f;
typedef __attribute__((ext_vector_type(8))) float v8f;

#define IMGS 96          // 32 batches * 3 channels
#define HW   512
#define W512 0.01227184630308513f   // 2*pi/512

// workspace layout (float offsets)
#define OFF_BTAB 0                               // 512*16 twiddle table
#define OFF_G    (512*16)                        // 96*512*8  row-DFT results
#define OFF_F    (OFF_G + IMGS*HW*8)             // 96*7*4*2  spectrum coeffs
#define OFF_H    (OFF_F + IMGS*7*4*2)            // 96*512*8  per-row recon coeffs
#define OFF_MAX  (OFF_H + IMGS*HW*8)             // 2 uints (max_lp, max_hp)

// ---------------- init: twiddle table + zero max slots ----------------
__global__ void k_init(float* __restrict__ ws) {
  int i = blockIdx.x * blockDim.x + threadIdx.x;     // 8192 threads
  if (i < 512 * 16) {
    int x = i >> 4, n = i & 15;
    float v = 0.f;
    int m = ((n & 3) * x) & 511;                     // exact integer angle reduction
    float ang = (float)m * W512;
    if (n < 4)      v = cosf(ang);                   // Re basis, kx = n
    else if (n < 8) v = -sinf(ang);                  // Im basis, kx = n-4
    ws[OFF_BTAB + i] = v;
  }
  if (i < 2) ((unsigned int*)(ws + OFF_MAX))[i] = 0u;
}

// ---------------- stage A: row DFT via WMMA f32 16x16x4 ----------------
// C(16 rows x 16 basis) = X_tile(16x512) * Btab(512x16), K swept 4 at a time.
__global__ void __launch_bounds__(32) k_dft_rows(const float* __restrict__ X,
                                                 float* __restrict__ ws) {
  const float* Btab = ws + OFF_BTAB;
  float* G = ws + OFF_G;
  int lane  = threadIdx.x;
  int img   = blockIdx.y;
  int y0    = blockIdx.x << 4;
  int n     = lane & 15;        // A: row M; B: col N; C: col N
  int khalf = lane >> 4;        // lanes 16-31 hold K=2,3 (A) / rows K=2,3 (B)
  const float* Xrow = X + ((size_t)img * HW + (y0 + n)) * HW;

  v8f c = {0.f, 0.f, 0.f, 0.f, 0.f, 0.f, 0.f, 0.f};
  for (int k0 = 0; k0 < HW; k0 += 4) {
    int kk = k0 + (khalf << 1);
    v2f a = *(const v2f*)(Xrow + kk);                // K=kk, kk+1 for row n
    v2f b;
    b.x = Btab[(kk << 4) + n];                       // B[K=kk  ][N=n]
    b.y = Btab[((kk + 1) << 4) + n];                 // B[K=kk+1][N=n]
    c = __builtin_amdgcn_wmma_f32_16x16x4_f32(false, a, false, b,
                                              (short)0, c, false, false);
  }
  if (n < 8) {                                       // only 8 basis cols are real
    #pragma unroll
    for (int i = 0; i < 8; ++i) {
      int y = y0 + i + (khalf << 3);                 // C: VGPR i -> M=i / M=8+i
      G[((size_t)img * HW + y) * 8 + n] = c[i];
    }
  }
}

// ---------------- stage B: column DFT -> F[ky=-3..3][kx=0..3] ----------------
__global__ void k_col_dft(float* __restrict__ ws) {
  __shared__ float red[14 * 256];
  const float* G = ws + OFF_G;
  float* F = ws + OFF_F;
  int kx = blockIdx.x, img = blockIdx.y, t = threadIdx.x;

  float acc[14];
  #pragma unroll
  for (int j = 0; j < 14; ++j) acc[j] = 0.f;

  for (int y = t; y < HW; y += 256) {
    float gre = G[((size_t)img * HW + y) * 8 + kx];
    float gim = G[((size_t)img * HW + y) * 8 + 4 + kx];
    float th = (float)y * W512;
    float c1, s1; __sincosf(th, &s1, &c1);
    float c2 = 2.f*c1*c1 - 1.f, s2 = 2.f*s1*c1;
    float c3 = 2.f*c1*c2 - c1,  s3 = 2.f*c1*s2 - s1;
    float cs[7] = { c3,  c2,  c1, 1.f, c1, c2, c3};
    float sn[7] = {-s3, -s2, -s1, 0.f, s1, s2, s3};
    #pragma unroll
    for (int j = 0; j < 7; ++j) {                    // e^{-i ky th}
      acc[j]     += gre * cs[j] + gim * sn[j];       // Fre
      acc[7 + j] += gim * cs[j] - gre * sn[j];       // Fim
    }
  }
  #pragma unroll
  for (int j = 0; j < 14; ++j) red[j * 256 + t] = acc[j];
  __syncthreads();
  for (int s = 128; s > 0; s >>= 1) {
    if (t < s)
      for (int j = 0; j < 14; ++j) red[j * 256 + t] += red[j * 256 + t + s];
    __syncthreads();
  }
  if (t < 14) {
    int kyi = t % 7, comp = t / 7;
    F[(((size_t)img * 7 + kyi) * 4 + kx) * 2 + comp] = red[t * 256];
  }
}

// ---------------- stage H: per-row reconstruction coefficients ----------------
__global__ void k_make_h(float* __restrict__ ws) {
  __shared__ float Fs[56];
  const float* F = ws + OFF_F;
  float* Hbuf = ws + OFF_H;
  int img = blockIdx.x, y = threadIdx.x;             // 512 threads
  if (y < 56) Fs[y] = F[(size_t)img * 56 + y];
  __syncthreads();

  float th = (float)y * W512;
  float c1, s1; __sincosf(th, &s1, &c1);
  float c2 = 2.f*c1*c1 - 1.f, s2 = 2.f*s1*c1;
  float c3 = 2.f*c1*c2 - c1,  s3 = 2.f*c1*s2 - s1;
  float cs[7] = { c3,  c2,  c1, 1.f, c1, c2, c3};
  float sn[7] = {-s3, -s2, -s1, 0.f, s1, s2, s3};

  #pragma unroll
  for (int kx = 0; kx < 4; ++kx) {
    float hre = 0.f, him = 0.f;
    int lo = (kx == 3) ? 1 : 0;                      // exclude (|ky|=3,|kx|=3)
    int hi = (kx == 3) ? 5 : 6;
    for (int kyi = lo; kyi <= hi; ++kyi) {
      float fre = Fs[(kyi * 4 + kx) * 2];
      float fim = Fs[(kyi * 4 + kx) * 2 + 1];
      float cc = cs[kyi], ss = sn[kyi];              // e^{+i ky th}
      hre += fre * cc - fim * ss;
      him += fre * ss + fim * cc;
    }
    Hbuf[((size_t)img * HW + y) * 8 + kx]     = hre;
    Hbuf[((size_t)img * HW + y) * 8 + 4 + kx] = him;
  }
}

// ---------------- per-pixel lp / hp ----------------
__device__ inline void lp_hp(const float* __restrict__ Hl, float xv, int x,
                             float& lpa, float& hpa) {
  float th = (float)x * W512;
  float c1, s1; __sincosf(th, &s1, &c1);
  float c2 = 2.f*c1*c1 - 1.f, s2 = 2.f*s1*c1;
  float c3 = 2.f*c1*c2 - c1,  s3 = 2.f*c1*s2 - s1;
  float lp = Hl[0] + 2.f * (Hl[1]*c1 - Hl[5]*s1 +
                            Hl[2]*c2 - Hl[6]*s2 +
                            Hl[3]*c3 - Hl[7]*s3);
  lp *= (1.f / (512.f * 512.f));
  lpa = fabsf(lp);
  hpa = fabsf(xv - lp);
}

// ---------------- pass 1: global maxima ----------------
__global__ void k_maxpass(const float* __restrict__ X, float* __restrict__ ws) {
  __shared__ float sl[256], sh[256];
  const float* Hbuf = ws + OFF_H;
  unsigned int* mx = (unsigned int*)(ws + OFF_MAX);
  int y = blockIdx.x, img = blockIdx.y, t = threadIdx.x;
  float Hl[8];
  const float* Hrow = Hbuf + ((size_t)img * HW + y) * 8;
  #pragma unroll
  for (int i = 0; i < 8; ++i) Hl[i] = Hrow[i];
  const float* Xrow = X + ((size_t)img * HW + y) * HW;

  float ml = 0.f, mh = 0.f;
  for (int x = t; x < HW; x += 256) {
    float lpa, hpa;
    lp_hp(Hl, Xrow[x], x, lpa, hpa);
    ml = fmaxf(ml, lpa); mh = fmaxf(mh, hpa);
  }
  sl[t] = ml; sh[t] = mh; __syncthreads();
  for (int s = 128; s > 0; s >>= 1) {
    if (t < s) { sl[t] = fmaxf(sl[t], sl[t + s]); sh[t] = fmaxf(sh[t], sh[t + s]); }
    __syncthreads();
  }
  if (t == 0) {                                      // values >= 0: uint order == float order
    atomicMax(&mx[0], __float_as_uint(sl[0]));
    atomicMax(&mx[1], __float_as_uint(sh[0]));
  }
}

// ---------------- pass 2: normalized writes + x copy ----------------
__global__ void k_write(const float* __restrict__ X, const float* __restrict__ ws,
                        float* __restrict__ out) {
  const float* Hbuf = ws + OFF_H;
  const unsigned int* mx = (const unsigned int*)(ws + OFF_MAX);
  int y = blockIdx.x, img = blockIdx.y, t = threadIdx.x;
  float inv_l = 1.f / __uint_as_float(mx[0]);
  float inv_h = 1.f / __uint_as_float(mx[1]);
  int b = img / 3, cin = img % 3;
  float Hl[8];
  const float* Hrow = Hbuf + ((size_t)img * HW + y) * 8;
  #pragma unroll
  for (int i = 0; i < 8; ++i) Hl[i] = Hrow[i];
  const float* Xrow = X + ((size_t)img * HW + y) * HW;
  size_t ob = (((size_t)b * 9 + cin)     * HW + y) * HW;
  size_t ol = (((size_t)b * 9 + 3 + cin) * HW + y) * HW;
  size_t oh = (((size_t)b * 9 + 6 + cin) * HW + y) * HW;

  for (int x = t; x < HW; x += 256) {
    float xv = Xrow[x];
    float lpa, hpa;
    lp_hp(Hl, xv, x, lpa, hpa);
    out[ob + x] = xv;
    out[ol + x] = lpa * inv_l;
    out[oh + x] = hpa * inv_h;
  }
}

extern "C" void kernel_launch(void* const* d_in, const int* in_sizes, int n_in,
                              void* d_out, int out_size, void* d_ws, size_t ws_size,
                              hipStream_t stream) {
  (void)in_sizes; (void)n_in; (void)out_size; (void)ws_size;
  const float* X = (const float*)d_in[0];
  float* out = (float*)d_out;
  float* ws  = (float*)d_ws;

  k_init<<<32, 256, 0, stream>>>(ws);

  dim3 gA(HW / 16, IMGS);                 // 32 x 96 tiles, 1 wave each
  k_dft_rows<<<gA, 32, 0, stream>>>(X, ws);

  dim3 gB(4, IMGS);
  k_col_dft<<<gB, 256, 0, stream>>>(ws);

  k_make_h<<<IMGS, HW, 0, stream>>>(ws);

  dim3 gP(HW, IMGS);
  k_maxpass<<<gP, 256, 0, stream>>>(X, ws);
  k_write<<<gP, 256, 0, stream>>>(X, ws, out);
}